// ProtoMIL_24584392802379
// MI455X (gfx1250) — compile-verified
//
#include <hip/hip_runtime.h>
#include <math.h>

// ---------------------------------------------------------------------------
// ProtoMIL forward for MI455X (gfx1250, wave32).
// Sizes fixed by the reference:
#define NROW 50000
#define DIM  2048
#define HDIM 256
#define CPROTO 64
#define CHID 16
#define NOUT 2
#define TOPK 10
#define EPSF 1e-6f

typedef __attribute__((ext_vector_type(2))) float v2f;
typedef __attribute__((ext_vector_type(4))) float v4f;
typedef __attribute__((ext_vector_type(8))) float v8f;

// ---------------------------------------------------------------------------
// Kernel 1: score_i = x_i . (w1 - w0)  (+ const bias diff added at write).
// One wave32 per row; 8 rows per 256-thread block. x streamed with NT hint.
__global__ __launch_bounds__(256)
void pm_score_kernel(const float* __restrict__ x,
                     const float* __restrict__ W_topk,
                     const float* __restrict__ b_topk,
                     float* __restrict__ scores) {
    __shared__ __align__(16) float wd[DIM];
    for (int i = threadIdx.x; i < DIM; i += 256)
        wd[i] = W_topk[DIM + i] - W_topk[i];
    __syncthreads();

    const int wave = threadIdx.x >> 5;
    const int lane = threadIdx.x & 31;
    const int row  = blockIdx.x * 8 + wave;
    if (row >= NROW) return;

    const v4f* __restrict__ xr = (const v4f*)(x + (size_t)row * DIM);
    const v4f* __restrict__ wr = (const v4f*)wd;

    v4f accv = {};
    #pragma unroll 4
    for (int i = lane; i < DIM / 4; i += 32) {
        v4f xv = __builtin_nontemporal_load(&xr[i]);   // stream, don't cache
        v4f wv = wr[i];                                // LDS
        accv += xv * wv;
    }
    float acc = accv.x + accv.y + accv.z + accv.w;
    #pragma unroll
    for (int off = 16; off > 0; off >>= 1)
        acc += __shfl_xor(acc, off, 32);
    if (lane == 0)
        scores[row] = acc + (b_topk[1] - b_topk[0]);
}

// ---------------------------------------------------------------------------
// Kernel 2: top-10 by score, ties broken toward lowest index (jax top_k
// stability). Single 1024-thread block, 10 sequential argmax passes.
__global__ __launch_bounds__(1024)
void pm_topk_kernel(float* __restrict__ scores, int* __restrict__ top_idx) {
    __shared__ float sv[1024];
    __shared__ int   si[1024];
    const int tid = threadIdx.x;
    for (int k = 0; k < TOPK; ++k) {
        float best = -INFINITY;
        int   bi   = 0x7fffffff;
        for (int i = tid; i < NROW; i += 1024) {
            float v = scores[i];
            if (v > best || (v == best && i < bi)) { best = v; bi = i; }
        }
        sv[tid] = best; si[tid] = bi;
        __syncthreads();
        for (int s = 512; s > 0; s >>= 1) {
            if (tid < s) {
                float ov = sv[tid + s]; int oi = si[tid + s];
                if (ov > sv[tid] || (ov == sv[tid] && oi < si[tid])) {
                    sv[tid] = ov; si[tid] = oi;
                }
            }
            __syncthreads();
        }
        if (tid == 0) {
            top_idx[k] = si[0];
            scores[si[0]] = -INFINITY;   // remove winner for next pass
        }
        __syncthreads();
    }
}

// ---------------------------------------------------------------------------
// Kernel 3: F[0:10]  = x_path[top_idx] @ W_metric^T + b  (f)
//           F[10:74] = prototype       @ W_metric^T + b  (p)
// Done as one 80x2048 * 2048x256 GEMM on V_WMMA_F32_16X16X4_F32.
// One wave per 16x16 output tile. Per the ISA VGPR layouts:
//   A 16x4 : lane<16 holds A[l16, {0,1}], lane>=16 holds A[l16, {2,3}]
//   B 4x16 : lane<16 holds B[{0,1}, l16], lane>=16 holds B[{2,3}, l16]
//   C 16x16: vgpr v, lane<16 -> D[v, l16]; lane>=16 -> D[v+8, l16]
__global__ __launch_bounds__(32)
void pm_metric_wmma_kernel(const float* __restrict__ x,
                           const float* __restrict__ proto,
                           const float* __restrict__ W_metric,
                           const float* __restrict__ b_metric,
                           const int* __restrict__ top_idx,
                           float* __restrict__ F) {
    const int mt   = blockIdx.x;          // 0..4  (row tiles, 80 padded rows)
    const int ht   = blockIdx.y;          // 0..15 (col tiles over HDIM=256)
    const int lane = threadIdx.x;         // 0..31
    const int half = lane >> 4;           // 0 or 1 -> K pair {0,1} vs {2,3}
    const int l16  = lane & 15;

    // A-row pointer for this lane (uniform per 16-lane half; clamp pad rows)
    int mrow = mt * 16 + l16;
    int mr   = (mrow < TOPK + CPROTO) ? mrow : (TOPK + CPROTO - 1);
    const float* __restrict__ arow =
        (mr < TOPK) ? (x + (size_t)top_idx[mr] * DIM)
                    : (proto + (size_t)(mr - TOPK) * DIM);
    const float* __restrict__ brow = W_metric + (size_t)(ht * 16 + l16) * DIM;

    v8f acc = {};
    #pragma unroll 4
    for (int d = 0; d < DIM; d += 4) {
        const float2 a2 = *(const float2*)(arow + d + 2 * half);
        const float2 b2 = *(const float2*)(brow + d + 2 * half);
        v2f a = { a2.x, a2.y };
        v2f b = { b2.x, b2.y };
        acc = __builtin_amdgcn_wmma_f32_16x16x4_f32(
            /*neg_a=*/false, a, /*neg_b=*/false, b,
            /*c_mod=*/(short)0, acc, /*reuse_a=*/false, /*reuse_b=*/false);
    }

    const int h  = ht * 16 + l16;
    const float bm = b_metric[h];
    #pragma unroll
    for (int v = 0; v < 8; ++v) {
        int m = mt * 16 + v + 8 * half;
        if (m < TOPK + CPROTO)
            F[(size_t)m * HDIM + h] = acc[v] + bm;
    }
}

// ---------------------------------------------------------------------------
// Kernel 4: pairwise distances, normalize, mean, rho->relu->cls->softmax.
__global__ __launch_bounds__(256)
void pm_finalize_kernel(const float* __restrict__ F,
                        const float* __restrict__ W_rho,
                        const float* __restrict__ b_rho,
                        const float* __restrict__ W_cls,
                        const float* __restrict__ b_cls,
                        float* __restrict__ out) {
    __shared__ float sf[TOPK * HDIM];        // 10 KB: the f rows
    __shared__ float sim[TOPK][CPROTO];
    __shared__ float rowmax[TOPK];
    __shared__ float simc[CPROTO];
    __shared__ float hidden[CHID];

    const int tid = threadIdx.x;
    for (int i = tid; i < TOPK * HDIM; i += 256) sf[i] = F[i];
    __syncthreads();

    for (int pair = tid; pair < TOPK * CPROTO; pair += 256) {
        const int t = pair / CPROTO, c = pair % CPROTO;
        const float* __restrict__ pr = F + (size_t)(TOPK + c) * HDIM;
        float s = 0.f;
        #pragma unroll 4
        for (int h = 0; h < HDIM; ++h) {
            float d = sf[t * HDIM + h] - pr[h] + EPSF;
            s += d * d;
        }
        sim[t][c] = sqrtf(s);
    }
    __syncthreads();

    if (tid < TOPK) {
        float m = sim[tid][0];
        for (int c = 1; c < CPROTO; ++c) m = fmaxf(m, sim[tid][c]);
        rowmax[tid] = m;
    }
    __syncthreads();

    if (tid < CPROTO) {
        float s = 0.f;
        for (int t = 0; t < TOPK; ++t) s += sim[t][tid] / rowmax[t];
        s *= (1.0f / TOPK);
        simc[tid] = s;
        out[5 + tid] = s;                    // sim_coding [64]
    }
    __syncthreads();

    if (tid < CHID) {
        float s = b_rho[tid];
        for (int c = 0; c < CPROTO; ++c) s += simc[c] * W_rho[tid * CPROTO + c];
        hidden[tid] = fmaxf(s, 0.f);
    }
    __syncthreads();

    if (tid == 0) {
        float l0 = b_cls[0], l1 = b_cls[1];
        for (int h = 0; h < CHID; ++h) {
            l0 += hidden[h] * W_cls[h];
            l1 += hidden[h] * W_cls[CHID + h];
        }
        out[0] = l0;                         // bag_logits
        out[1] = l1;
        float mx = fmaxf(l0, l1);
        float e0 = expf(l0 - mx), e1 = expf(l1 - mx);
        float inv = 1.f / (e0 + e1);
        out[2] = e0 * inv;                   // Y_prob
        out[3] = e1 * inv;
        out[4] = (l1 > l0) ? 1.f : 0.f;      // Y_hat (argmax, ties -> 0)
    }
}

// ---------------------------------------------------------------------------
extern "C" void kernel_launch(void* const* d_in, const int* in_sizes, int n_in,
                              void* d_out, int out_size, void* d_ws, size_t ws_size,
                              hipStream_t stream) {
    const float* x_path   = (const float*)d_in[0];
    const float* proto    = (const float*)d_in[1];
    const float* W_topk   = (const float*)d_in[2];
    const float* b_topk   = (const float*)d_in[3];
    const float* W_metric = (const float*)d_in[4];
    const float* b_metric = (const float*)d_in[5];
    const float* W_rho    = (const float*)d_in[6];
    const float* b_rho    = (const float*)d_in[7];
    const float* W_cls    = (const float*)d_in[8];
    const float* b_cls    = (const float*)d_in[9];
    float* out = (float*)d_out;

    // Workspace layout (all rewritten every call -> deterministic):
    //   [0, 200704)          : scores float[50000] (padded)
    //   [200704, 200960)     : top_idx int[10] (padded to 256 B)
    //   [200960, 282880)     : F float[80*256]
    char* ws = (char*)d_ws;
    float* scores  = (float*)ws;
    int*   top_idx = (int*)(ws + 200704);
    float* F       = (float*)(ws + 200960);

    pm_score_kernel<<<dim3((NROW + 7) / 8), dim3(256), 0, stream>>>(
        x_path, W_topk, b_topk, scores);

    pm_topk_kernel<<<dim3(1), dim3(1024), 0, stream>>>(scores, top_idx);

    pm_metric_wmma_kernel<<<dim3(5, 16), dim3(32), 0, stream>>>(
        x_path, proto, W_metric, b_metric, top_idx, F);

    pm_finalize_kernel<<<dim3(1), dim3(256), 0, stream>>>(
        F, W_rho, b_rho, W_cls, b_cls, out);
}